// AFNO3DAdapter_14259291423357
// MI455X (gfx1250) — compile-verified
//
#include <hip/hip_runtime.h>
#include <math.h>

typedef __attribute__((ext_vector_type(2))) float v2f;
typedef __attribute__((ext_vector_type(8))) float v8f;

#define TPB 256
#define WAVES 8

// ---------------------------------------------------------------------------
// WMMA helpers (V_WMMA_F32_16X16X4_F32, wave32)
// A: 16x4 (M x K)  -> 2 VGPRs: lane m=lane&15, lane-half h: regs hold K=2h, 2h+1
// B: 4x16 (K x N)  -> 2 VGPRs: lane n=lane&15, lane-half h: regs hold K=2h, 2h+1
// C/D: 16x16       -> 8 VGPRs: reg r, lanes0-15 -> M=r, lanes16-31 -> M=r+8; N=lane&15
// ---------------------------------------------------------------------------
__device__ __forceinline__ v8f wmma4(v2f a, v2f b, v8f c) {
  return __builtin_amdgcn_wmma_f32_16x16x4_f32(false, a, false, b, (short)0, c,
                                               false, false);
}

__device__ __forceinline__ v8f zero8() {
  v8f z = {0.f, 0.f, 0.f, 0.f, 0.f, 0.f, 0.f, 0.f};
  return z;
}

// A fragment from row-major A[m][k] (lda), tile row base m0, K base k0
__device__ __forceinline__ v2f afrag(const float* A, int lda, int m0, int k0,
                                     int lane) {
  int m = m0 + (lane & 15);
  int k = k0 + ((lane >> 4) << 1);
  v2f r;
  r.x = A[m * lda + k];
  r.y = A[m * lda + k + 1];
  return r;
}

// B fragment from row-major B[k][n] (ldb), K base k0, N base n0
__device__ __forceinline__ v2f bfrag(const float* B, int ldb, int k0, int n0,
                                     int lane) {
  int n = n0 + (lane & 15);
  int k = k0 + ((lane >> 4) << 1);
  v2f r;
  r.x = B[k * ldb + n];
  r.y = B[(k + 1) * ldb + n];
  return r;
}

__device__ __forceinline__ float gelu_exact(float v) {
  return 0.5f * v * (1.0f + erff(v * 0.7071067811865475f));
}

// ---------------------------------------------------------------------------
// Twiddle matrices (ortho norm: 1/8 per axis per direction)
//  WL  [16][64]: rows 0-7 = cos(2pi kl l/64)/8, rows 8-15 = -sin(...)/8 (fwd l)
//  W32 [32][64]: fwd h/w DFT, e^{-i}
//  V64 [64][32]: inv h/w DFT, e^{+i}
//  VL  [64][16]: cols 0-7 = eps*cos/8, cols 8-15 = -eps*sin/8 (inv l, real out)
// ---------------------------------------------------------------------------
__global__ void k_twiddle(float* WL, float* W32r, float* W32i, float* V64r,
                          float* V64i, float* VL) {
  const float s = 0.125f;
  const float w = 6.283185307179586f / 64.0f;
  for (int idx = threadIdx.x; idx < 16 * 64; idx += TPB) {
    int row = idx >> 6, l = idx & 63;
    int kl = row & 7;
    float th = w * (float)((kl * l) & 63);
    WL[idx] = (row < 8) ? s * cosf(th) : -s * sinf(th);
  }
  for (int idx = threadIdx.x; idx < 32 * 64; idx += TPB) {
    int k = idx >> 6, j = idx & 63;
    float th = w * (float)((k * j) & 63);
    W32r[idx] = s * cosf(th);
    W32i[idx] = -s * sinf(th);
  }
  for (int idx = threadIdx.x; idx < 64 * 32; idx += TPB) {
    int h = idx >> 5, k = idx & 31;
    float th = w * (float)((k * h) & 63);
    V64r[idx] = s * cosf(th);
    V64i[idx] = s * sinf(th);
  }
  for (int idx = threadIdx.x; idx < 64 * 16; idx += TPB) {
    int l = idx >> 4, col = idx & 15;
    int kl = col & 7;
    float eps = (kl == 0) ? 1.0f : 2.0f;
    float th = w * (float)((kl * l) & 63);
    VL[idx] = (col < 8) ? eps * s * cosf(th) : -eps * s * sinf(th);
  }
}

// ---------------------------------------------------------------------------
// Forward DFT along l: per (b,h,w) slab [64 l][128 c] -> [8 kl][128 c] cplx
// One GEMM M=16 ([Re;Im]), K=64, N=128 (8 waves x one 16-wide N tile)
// ---------------------------------------------------------------------------
__global__ void k_fwd_l(const float* __restrict__ x,
                        const float* __restrict__ WL, float* __restrict__ Yr,
                        float* __restrict__ Yi) {
  __shared__ float lA[16 * 64];
  for (int i = threadIdx.x; i < 1024; i += TPB) lA[i] = WL[i];
  __syncthreads();
  const int lane = threadIdx.x & 31;
  const int wv = threadIdx.x >> 5;
  const size_t slab = blockIdx.x;  // (b,h,w)
  const float* xs = x + slab * 8192;
  const int cbase = wv * 16;
  v8f acc = zero8();
#pragma unroll
  for (int ks = 0; ks < 16; ++ks) {
    int k0 = ks * 4;
    v2f a = afrag(lA, 64, 0, k0, lane);
    v2f b = bfrag(xs, 128, k0, cbase, lane);
    acc = wmma4(a, b, acc);
  }
  const int n = lane & 15;
  const int mo = (lane >> 4) << 3;  // 0: rows 0-7 (real) | 8: rows 8-15 (imag)
  float* outp = (mo == 0) ? Yr : Yi;
  size_t base = slab * 1024 + (size_t)cbase + n;
#pragma unroll
  for (int r = 0; r < 8; ++r) {
    outp[base + (size_t)r * 128] = acc[r];
  }
}

// ---------------------------------------------------------------------------
// Complex GEMM stage: C[batch][m][n] = A[m][k] (*) B[batch][k][n]
//   Cr = Ar*Br - Ai*Bi ; Ci = Ar*Bi + Ai*Br     (A in LDS, minus via -Bi frag)
// ---------------------------------------------------------------------------
template <int M, int K>
__global__ void k_cgemm(const float* __restrict__ Ar,
                        const float* __restrict__ Ai,
                        const float* __restrict__ Br,
                        const float* __restrict__ Bi, float* __restrict__ Cr,
                        float* __restrict__ Ci, int N, long inStride,
                        long outStride) {
  __shared__ float lAr[M * K];
  __shared__ float lAi[M * K];
  for (int i = threadIdx.x; i < M * K; i += TPB) {
    lAr[i] = Ar[i];
    lAi[i] = Ai[i];
  }
  __syncthreads();
  const int lane = threadIdx.x & 31;
  const int wv = threadIdx.x >> 5;
  const int batch = blockIdx.y;
  const int nbase = ((int)blockIdx.x * WAVES + wv) * 16;
  const float* br_ = Br + (long)batch * inStride;
  const float* bi_ = Bi + (long)batch * inStride;
  constexpr int MT = M / 16;
  v8f accR[MT], accI[MT];
#pragma unroll
  for (int t = 0; t < MT; ++t) {
    accR[t] = zero8();
    accI[t] = zero8();
  }
#pragma unroll
  for (int ks = 0; ks < K / 4; ++ks) {
    int k0 = ks * 4;
    v2f br = bfrag(br_, N, k0, nbase, lane);
    v2f bi = bfrag(bi_, N, k0, nbase, lane);
    v2f bin;
    bin.x = -bi.x;
    bin.y = -bi.y;
#pragma unroll
    for (int t = 0; t < MT; ++t) {
      v2f ar = afrag(lAr, K, t * 16, k0, lane);
      v2f ai = afrag(lAi, K, t * 16, k0, lane);
      accR[t] = wmma4(ar, br, accR[t]);
      accR[t] = wmma4(ai, bin, accR[t]);
      accI[t] = wmma4(ar, bi, accI[t]);
      accI[t] = wmma4(ai, br, accI[t]);
    }
  }
  const int n = lane & 15;
  const int mo = (lane >> 4) << 3;
  long ob = (long)batch * outStride + nbase + n;
#pragma unroll
  for (int t = 0; t < MT; ++t) {
#pragma unroll
    for (int r = 0; r < 8; ++r) {
      long row = t * 16 + mo + r;
      Cr[ob + row * (long)N] = accR[t][r];
      Ci[ob + row * (long)N] = accI[t][r];
    }
  }
}

// ---------------------------------------------------------------------------
// Spectral per-block MLP + FiLM adapters. One wave owns one 16-position tile.
// tile: LDS [16 pos][16 ch]; hbuf: LDS [16][32]. All waves run uniformly.
// ---------------------------------------------------------------------------
__device__ void adapter(float* tile, float* hbuf, const float* dw,
                        const float* db, const float* fw, const float* fb,
                        int lane) {
  const int n = lane & 15;
  const int mo = (lane >> 4) << 3;
  // h = gelu(v * dw + db)  [16 x 32]
  v8f h0 = zero8(), h1 = zero8();
#pragma unroll
  for (int ks = 0; ks < 4; ++ks) {
    int k0 = ks * 4;
    v2f a = afrag(tile, 16, 0, k0, lane);
    h0 = wmma4(a, bfrag(dw, 32, k0, 0, lane), h0);
    h1 = wmma4(a, bfrag(dw, 32, k0, 16, lane), h1);
  }
  float bias0 = db[n], bias1 = db[16 + n];
#pragma unroll
  for (int r = 0; r < 8; ++r) {
    hbuf[(mo + r) * 32 + n] = gelu_exact(h0[r] + bias0);
    hbuf[(mo + r) * 32 + 16 + n] = gelu_exact(h1[r] + bias1);
  }
  __syncthreads();
  // film = h * fw + fb ; gamma = cols 0-15, beta = cols 16-31
  v8f f0 = zero8(), f1 = zero8();
#pragma unroll
  for (int ks = 0; ks < 8; ++ks) {
    int k0 = ks * 4;
    v2f a = afrag(hbuf, 32, 0, k0, lane);
    f0 = wmma4(a, bfrag(fw, 32, k0, 0, lane), f0);
    f1 = wmma4(a, bfrag(fw, 32, k0, 16, lane), f1);
  }
  float gb = fb[n], bb = fb[16 + n];
  __syncthreads();
#pragma unroll
  for (int r = 0; r < 8; ++r) {
    float v = tile[(mo + r) * 16 + n];
    tile[(mo + r) * 16 + n] = v * (1.0f + f0[r] + gb) + (f1[r] + bb);
  }
  __syncthreads();
}

__device__ void cmlp(float* tR, float* tI, const float* wr, const float* wi,
                     const float* br, const float* bi, int lane, bool doGelu) {
  const int n = lane & 15;
  const int mo = (lane >> 4) << 3;
  v8f oR = zero8(), oI = zero8();
#pragma unroll
  for (int ks = 0; ks < 4; ++ks) {
    int k0 = ks * 4;
    v2f aR = afrag(tR, 16, 0, k0, lane);
    v2f aI = afrag(tI, 16, 0, k0, lane);
    v2f bR = bfrag(wr, 16, k0, 0, lane);
    v2f bI = bfrag(wi, 16, k0, 0, lane);
    v2f bIn;
    bIn.x = -bI.x;
    bIn.y = -bI.y;
    oR = wmma4(aR, bR, oR);
    oR = wmma4(aI, bIn, oR);
    oI = wmma4(aI, bR, oI);
    oI = wmma4(aR, bI, oI);
  }
  float bbr = br[n], bbi = bi[n];
  __syncthreads();
#pragma unroll
  for (int r = 0; r < 8; ++r) {
    float vr = oR[r] + bbr, vi = oI[r] + bbi;
    if (doGelu) {
      vr = gelu_exact(vr);
      vi = gelu_exact(vi);
    }
    tR[(mo + r) * 16 + n] = vr;
    tI[(mo + r) * 16 + n] = vi;
  }
  __syncthreads();
}

__global__ void k_spectral(
    float* __restrict__ Xr, float* __restrict__ Xi, const float* w1,
    const float* b1, const float* w2, const float* b2, const float* in_dw,
    const float* in_db, const float* in_fw, const float* in_fb,
    const float* mid_dw, const float* mid_db, const float* mid_fw,
    const float* mid_fb, const float* out_dw, const float* out_db,
    const float* out_fw, const float* out_fb) {
  __shared__ float sTR[WAVES][256];
  __shared__ float sTI[WAVES][256];
  __shared__ float sH[WAVES][512];
  const int lane = threadIdx.x & 31;
  const int wv = threadIdx.x >> 5;
  float* tR = sTR[wv];
  float* tI = sTI[wv];
  float* hb = sH[wv];
  const size_t p0 = ((size_t)blockIdx.x * WAVES + wv) * 16;

  for (int nb = 0; nb < 8; ++nb) {
    for (int i = lane; i < 256; i += 32) {
      int p = i >> 4, ch = i & 15;
      size_t g = (p0 + p) * 128 + (size_t)nb * 16 + ch;
      tR[i] = Xr[g];
      tI[i] = Xi[g];
    }
    __syncthreads();
    adapter(tR, hb, in_dw + nb * 512, in_db + nb * 32, in_fw + nb * 1024,
            in_fb + nb * 32, lane);
    adapter(tI, hb, in_dw + nb * 512, in_db + nb * 32, in_fw + nb * 1024,
            in_fb + nb * 32, lane);
    cmlp(tR, tI, w1 + nb * 256, w1 + 2048 + nb * 256, b1 + nb * 16,
         b1 + 128 + nb * 16, lane, true);
    adapter(tR, hb, mid_dw + nb * 512, mid_db + nb * 32, mid_fw + nb * 1024,
            mid_fb + nb * 32, lane);
    adapter(tI, hb, mid_dw + nb * 512, mid_db + nb * 32, mid_fw + nb * 1024,
            mid_fb + nb * 32, lane);
    cmlp(tR, tI, w2 + nb * 256, w2 + 2048 + nb * 256, b2 + nb * 16,
         b2 + 128 + nb * 16, lane, false);
    adapter(tR, hb, out_dw + nb * 512, out_db + nb * 32, out_fw + nb * 1024,
            out_fb + nb * 32, lane);
    adapter(tI, hb, out_dw + nb * 512, out_db + nb * 32, out_fw + nb * 1024,
            out_fb + nb * 32, lane);
    for (int i = lane; i < 256; i += 32) {
      int p = i >> 4, ch = i & 15;
      size_t g = (p0 + p) * 128 + (size_t)nb * 16 + ch;
      Xr[g] = tR[i];
      Xi[g] = tI[i];
    }
    __syncthreads();
  }
}

// ---------------------------------------------------------------------------
// Inverse-l (real output, Hermitian eps-weights folded into VL) + residual
// Per (b,h,w) slab: out[64 l][128 c] = VL[64][16] x [Zr(8);Zi(8)][128] + x
// ---------------------------------------------------------------------------
__global__ void k_fin(const float* __restrict__ Zr, const float* __restrict__ Zi,
                      const float* __restrict__ VL, const float* __restrict__ x,
                      float* __restrict__ out) {
  __shared__ float lA[64 * 16];
  for (int i = threadIdx.x; i < 1024; i += TPB) lA[i] = VL[i];
  __syncthreads();
  const int lane = threadIdx.x & 31;
  const int wv = threadIdx.x >> 5;
  const size_t slab = blockIdx.x;
  const int cbase = wv * 16;
  const float* zr = Zr + slab * 1024;
  const float* zi = Zi + slab * 1024;
  v8f acc[4];
#pragma unroll
  for (int t = 0; t < 4; ++t) acc[t] = zero8();
  const int nb = cbase + (lane & 15);
#pragma unroll
  for (int ks = 0; ks < 4; ++ks) {
    int k0 = ks * 4;
    int kk = k0 + ((lane >> 4) << 1);
    v2f b;
    b.x = (kk < 8) ? zr[kk * 128 + nb] : zi[(kk - 8) * 128 + nb];
    b.y = (kk + 1 < 8) ? zr[(kk + 1) * 128 + nb] : zi[(kk - 7) * 128 + nb];
#pragma unroll
    for (int t = 0; t < 4; ++t) {
      v2f a = afrag(lA, 16, t * 16, k0, lane);
      acc[t] = wmma4(a, b, acc[t]);
    }
  }
  const int n = lane & 15;
  const int mo = (lane >> 4) << 3;
  size_t ob = slab * 8192 + (size_t)cbase + n;
#pragma unroll
  for (int t = 0; t < 4; ++t) {
#pragma unroll
    for (int r = 0; r < 8; ++r) {
      size_t o = ob + (size_t)(t * 16 + mo + r) * 128;
      out[o] = acc[t][r] + x[o];
    }
  }
}

// ---------------------------------------------------------------------------
// Host launcher
// ---------------------------------------------------------------------------
extern "C" void kernel_launch(void* const* d_in, const int* in_sizes, int n_in,
                              void* d_out, int out_size, void* d_ws,
                              size_t ws_size, hipStream_t stream) {
  const float* x = (const float*)d_in[0];
  const float* w1 = (const float*)d_in[1];
  const float* b1 = (const float*)d_in[2];
  const float* w2 = (const float*)d_in[3];
  const float* b2 = (const float*)d_in[4];
  const float* in_dw = (const float*)d_in[5];
  const float* in_db = (const float*)d_in[6];
  const float* in_fw = (const float*)d_in[7];
  const float* in_fb = (const float*)d_in[8];
  const float* mid_dw = (const float*)d_in[9];
  const float* mid_db = (const float*)d_in[10];
  const float* mid_fw = (const float*)d_in[11];
  const float* mid_fb = (const float*)d_in[12];
  const float* out_dw = (const float*)d_in[13];
  const float* out_db = (const float*)d_in[14];
  const float* out_fw = (const float*)d_in[15];
  const float* out_fb = (const float*)d_in[16];
  float* out = (float*)d_out;

  float* ws = (float*)d_ws;
  float* WL = ws;              // 1024
  float* W32r = WL + 1024;     // 2048
  float* W32i = W32r + 2048;   // 2048
  float* V64r = W32i + 2048;   // 2048
  float* V64i = V64r + 2048;   // 2048
  float* VL = V64i + 2048;     // 1024
  float* Y1r = VL + 1024;      // 8388608  (reused as Z1r)
  float* Y1i = Y1r + 8388608;  // 8388608  (reused as Z1i)
  float* Y2r = Y1i + 8388608;  // 4194304  (reused as Z2r)
  float* Y2i = Y2r + 4194304;  // 4194304  (reused as Z2i)
  float* Y3r = Y2i + 4194304;  // 2097152
  float* Y3i = Y3r + 2097152;  // 2097152
  float* Z2r = Y2r;
  float* Z2i = Y2i;
  float* Z1r = Y1r;
  float* Z1i = Y1i;

  k_twiddle<<<1, TPB, 0, stream>>>(WL, W32r, W32i, V64r, V64i, VL);

  // forward l-DFT: x[2,64,64,64,128] -> Y1[2,64,64,8,128] cplx
  k_fwd_l<<<8192, TPB, 0, stream>>>(x, WL, Y1r, Y1i);

  // forward w-DFT: batch=(b,h)=128, K=64(w), N=1024(kl*c), M=32(kw)
  k_cgemm<32, 64><<<dim3(8, 128), TPB, 0, stream>>>(
      W32r, W32i, Y1r, Y1i, Y2r, Y2i, 1024, 65536L, 32768L);

  // forward h-DFT: batch=b=2, K=64(h), N=32768(kw*kl*c), M=32(kh)
  k_cgemm<32, 64><<<dim3(256, 2), TPB, 0, stream>>>(
      W32r, W32i, Y2r, Y2i, Y3r, Y3i, 32768, 2097152L, 1048576L);

  // spectral adapters + complex block MLP (in-place on Y3)
  k_spectral<<<128, TPB, 0, stream>>>(Y3r, Y3i, w1, b1, w2, b2, in_dw, in_db,
                                      in_fw, in_fb, mid_dw, mid_db, mid_fw,
                                      mid_fb, out_dw, out_db, out_fw, out_fb);

  // inverse h: batch=2, K=32(kh), N=32768, M=64(h)
  k_cgemm<64, 32><<<dim3(256, 2), TPB, 0, stream>>>(
      V64r, V64i, Y3r, Y3i, Z2r, Z2i, 32768, 1048576L, 2097152L);

  // inverse w: batch=(b,h)=128, K=32(kw), N=1024, M=64(w)
  k_cgemm<64, 32><<<dim3(8, 128), TPB, 0, stream>>>(
      V64r, V64i, Z2r, Z2i, Z1r, Z1i, 1024, 32768L, 65536L);

  // inverse l (real) + residual
  k_fin<<<8192, TPB, 0, stream>>>(Z1r, Z1i, VL, x, out);

  (void)in_sizes;
  (void)n_in;
  (void)out_size;
  (void)ws_size;
}